// MultiLayerGCN_49185965473826
// MI455X (gfx1250) — compile-verified
//
#include <hip/hip_runtime.h>
#include <math.h>

typedef __attribute__((ext_vector_type(2))) float v2f;
typedef __attribute__((ext_vector_type(8))) float v8f;

#define N_NODES   50000
#define N_EDGES   600000
#define N_GRAPHS  128

// ---------------------------------------------------------------------------
// Utility: zero a float buffer
// ---------------------------------------------------------------------------
__global__ void gcn_zero_f32(float* __restrict__ p, int n) {
  int i = blockIdx.x * blockDim.x + threadIdx.x;
  if (i < n) p[i] = 0.0f;
}

// ---------------------------------------------------------------------------
// Degree count: deg[dst] += 1 per edge (self loop +1 folded in later)
// ---------------------------------------------------------------------------
__global__ void gcn_deg_count(const int* __restrict__ dst, float* __restrict__ deg, int nE) {
  int e = blockIdx.x * blockDim.x + threadIdx.x;
  if (e < nE) unsafeAtomicAdd(&deg[dst[e]], 1.0f);
}

// dinv[i] = rsqrt(deg_edges[i] + 1)   (self-loop makes deg >= 1 always)
__global__ void gcn_finish_dinv(float* __restrict__ dinv, int n) {
  int i = blockIdx.x * blockDim.x + threadIdx.x;
  if (i < n) dinv[i] = rsqrtf(dinv[i] + 1.0f);
}

// norm[e] = dinv[src[e]] * dinv[dst[e]]
__global__ void gcn_edge_norm(const int* __restrict__ src, const int* __restrict__ dst,
                              const float* __restrict__ dinv, float* __restrict__ nrm, int nE) {
  int e = blockIdx.x * blockDim.x + threadIdx.x;
  if (e < nE) nrm[e] = dinv[src[e]] * dinv[dst[e]];
}

// ---------------------------------------------------------------------------
// GEMM via V_WMMA_F32_16X16X4_F32.  H[M,OUT] = X[M,IN] @ W[IN,OUT]
//
// Each wave computes an 80x32 output tile = 5 (M) x 2 (N) WMMA subtiles.
// Per K-step of 4: 2 B frags (4 floats) + 5 A frags (10 floats) feed 10
// WMMAs -> ~1.4 lane-loads per WMMA (vs 4 for a single-tile wave).
// 50000 = 80*625, so the M dimension tiles exactly.
//
// Fragment layouts (wave32, per ISA 7.12.2):
//  A (16x4 f32, 2 VGPRs): lane l: M = l&15, K = v + (l>>4)*2
//  B (4x16 f32, 2 VGPRs): lane l: N = l&15, K = v + (l>>4)*2
//  C/D (16x16 f32, 8 VGPRs): lane l: N = l&15, M = v + (l>>4)*8
// ---------------------------------------------------------------------------
template <int IN, int OUT>
__global__ void gcn_gemm_wmma(const float* __restrict__ X, const float* __restrict__ W,
                              float* __restrict__ H) {
  const int m0   = blockIdx.x * 80;
  const int n0   = blockIdx.y * 32;
  const int lane = threadIdx.x;        // 0..31 (wave32)
  const int half = lane >> 4;          // 0 or 1
  const int lid  = lane & 15;

  const float* xbase = X + (size_t)(m0 + lid) * IN;  // row for M-subtile 0

  v8f acc[5][2];
#pragma unroll
  for (int mi = 0; mi < 5; ++mi)
#pragma unroll
    for (int ni = 0; ni < 2; ++ni)
      acc[mi][ni] = (v8f){};

#pragma unroll 2
  for (int k0 = 0; k0 < IN; k0 += 4) {
    const int ka = k0 + half * 2;

    v2f b[2];
#pragma unroll
    for (int ni = 0; ni < 2; ++ni) {
      b[ni].x = W[(size_t)(ka + 0) * OUT + n0 + ni * 16 + lid];
      b[ni].y = W[(size_t)(ka + 1) * OUT + n0 + ni * 16 + lid];
    }

#pragma unroll
    for (int mi = 0; mi < 5; ++mi) {
      v2f a;
      a.x = xbase[(size_t)mi * 16 * IN + ka];
      a.y = xbase[(size_t)mi * 16 * IN + ka + 1];
#pragma unroll
      for (int ni = 0; ni < 2; ++ni)
        acc[mi][ni] = __builtin_amdgcn_wmma_f32_16x16x4_f32(
            /*neg_a=*/false, a, /*neg_b=*/false, b[ni],
            /*c_mod=*/(short)0, acc[mi][ni], /*reuse_a=*/false, /*reuse_b=*/false);
    }
  }

#pragma unroll
  for (int mi = 0; mi < 5; ++mi)
#pragma unroll
    for (int ni = 0; ni < 2; ++ni)
#pragma unroll
      for (int v = 0; v < 8; ++v)
        H[(size_t)(m0 + mi * 16 + v + half * 8) * OUT + n0 + ni * 16 + lid] =
            acc[mi][ni][v];
}

// ---------------------------------------------------------------------------
// Edge scatter: agg[dst] += norm[e] * H[src]   (one wave per edge)
// ---------------------------------------------------------------------------
template <int OUT>
__global__ void gcn_edge_scatter(const float* __restrict__ H, const int* __restrict__ src,
                                 const int* __restrict__ dst, const float* __restrict__ nrm,
                                 float* __restrict__ agg, int nE) {
  int gid  = blockIdx.x * blockDim.x + threadIdx.x;
  int e    = gid >> 5;
  int lane = gid & 31;
  if (e >= nE) return;
  int   s = src[e];
  int   d = dst[e];
  float w = nrm[e];
  if constexpr (OUT == 128) {
    float4 v = ((const float4*)(H + (size_t)s * OUT))[lane];
    float* op = agg + (size_t)d * OUT + lane * 4;
    unsafeAtomicAdd(op + 0, v.x * w);
    unsafeAtomicAdd(op + 1, v.y * w);
    unsafeAtomicAdd(op + 2, v.z * w);
    unsafeAtomicAdd(op + 3, v.w * w);
  } else {  // OUT == 64
    float2 v = ((const float2*)(H + (size_t)s * OUT))[lane];
    float* op = agg + (size_t)d * OUT + lane * 2;
    unsafeAtomicAdd(op + 0, v.x * w);
    unsafeAtomicAdd(op + 1, v.y * w);
  }
}

// ---------------------------------------------------------------------------
// Finalize layer: agg = [relu](agg + dinv[i]^2 * H[i] + bias)
// (self-loop contribution folded in here: norm_self = dinv*dinv)
// ---------------------------------------------------------------------------
template <int OUT, bool RELU>
__global__ void gcn_finalize(float* __restrict__ agg, const float* __restrict__ H,
                             const float* __restrict__ dinv, const float* __restrict__ bias,
                             int total) {
  int idx = blockIdx.x * blockDim.x + threadIdx.x;
  if (idx >= total) return;
  int node = idx / OUT;
  int f    = idx - node * OUT;
  float di = dinv[node];
  float v  = agg[idx] + di * di * H[idx] + bias[f];
  if constexpr (RELU) v = fmaxf(v, 0.0f);
  agg[idx] = v;
}

// ---------------------------------------------------------------------------
// Global mean pool accumulation (wave per node; 64 feats -> 2 per lane)
// ---------------------------------------------------------------------------
__global__ void gcn_pool(const float* __restrict__ h, const int* __restrict__ batch,
                         float* __restrict__ sums, float* __restrict__ cnts, int n) {
  int gid  = blockIdx.x * blockDim.x + threadIdx.x;
  int node = gid >> 5;
  int lane = gid & 31;
  if (node >= n) return;
  int g = batch[node];
  float2 v = ((const float2*)(h + (size_t)node * 64))[lane];
  unsafeAtomicAdd(&sums[g * 64 + lane * 2 + 0], v.x);
  unsafeAtomicAdd(&sums[g * 64 + lane * 2 + 1], v.y);
  if (lane == 0) unsafeAtomicAdd(&cnts[g], 1.0f);
}

// ---------------------------------------------------------------------------
// Per-graph mean + softmax over 64 classes (one thread per graph)
// ---------------------------------------------------------------------------
__global__ void gcn_softmax(const float* __restrict__ sums, const float* __restrict__ cnts,
                            float* __restrict__ out) {
  int g = threadIdx.x;
  if (g >= N_GRAPHS) return;
  float inv = 1.0f / fmaxf(cnts[g], 1.0f);
  float p[64];
  float m = -1e30f;
#pragma unroll
  for (int f = 0; f < 64; ++f) {
    p[f] = sums[g * 64 + f] * inv;
    m = fmaxf(m, p[f]);
  }
  float s = 0.0f;
#pragma unroll
  for (int f = 0; f < 64; ++f) {
    p[f] = __expf(p[f] - m);
    s += p[f];
  }
  float r = 1.0f / s;
#pragma unroll
  for (int f = 0; f < 64; ++f) out[g * 64 + f] = p[f] * r;
}

// ---------------------------------------------------------------------------
// Host launcher
// ---------------------------------------------------------------------------
extern "C" void kernel_launch(void* const* d_in, const int* in_sizes, int n_in,
                              void* d_out, int out_size, void* d_ws, size_t ws_size,
                              hipStream_t stream) {
  const float* x     = (const float*)d_in[0];
  const int*   eidx  = (const int*)d_in[1];
  const int*   batch = (const int*)d_in[2];
  const float* W0    = (const float*)d_in[3];
  const float* b0    = (const float*)d_in[4];
  const float* W1    = (const float*)d_in[5];
  const float* b1    = (const float*)d_in[6];
  const float* W2    = (const float*)d_in[7];
  const float* b2    = (const float*)d_in[8];
  float*       out   = (float*)d_out;

  const int* src = eidx;             // edge_index[0]
  const int* dst = eidx + N_EDGES;   // edge_index[1]

  // ---- workspace carve-out (floats) ----
  float* ws   = (float*)d_ws;
  float* dinv = ws;                       // 50048   (50000, padded)
  float* nrm  = ws + 50048;               // 600064  (600000, padded)
  float* bufA = ws + 650112;              // 6,400,000 (50000x128)
  float* bufB = ws + 7050112;             // 6,400,000 (50000x128)
  float* sums = ws + 13450112;            // 8192 (128x64)
  float* cnts = ws + 13458304;            // 128

  const int B = 256;
  const int nodeBlocks  = (N_NODES + B - 1) / B;
  const int edgeBlocks  = (N_EDGES + B - 1) / B;
  const int scatBlocks  = (N_EDGES * 32 + B - 1) / B;   // wave per edge
  const int poolBlocks  = (N_NODES * 32 + B - 1) / B;   // wave per node
  const int feat128     = N_NODES * 128;
  const int feat64      = N_NODES * 64;
  const int fin128Blk   = (feat128 + B - 1) / B;
  const int fin64Blk    = (feat64 + B - 1) / B;

  // ---- normalization precompute ----
  gcn_zero_f32<<<nodeBlocks, B, 0, stream>>>(dinv, N_NODES);
  gcn_zero_f32<<<(8192 + B - 1) / B, B, 0, stream>>>(sums, 8192);
  gcn_zero_f32<<<1, B, 0, stream>>>(cnts, N_GRAPHS);
  gcn_deg_count<<<edgeBlocks, B, 0, stream>>>(dst, dinv, N_EDGES);
  gcn_finish_dinv<<<nodeBlocks, B, 0, stream>>>(dinv, N_NODES);
  gcn_edge_norm<<<edgeBlocks, B, 0, stream>>>(src, dst, dinv, nrm, N_EDGES);

  dim3 g128(N_NODES / 80, 128 / 32);  // (625, 4)
  dim3 g64 (N_NODES / 80,  64 / 32);  // (625, 2)

  // ---- layer 0: h = relu(scatter(norm * (x@W0)[src]) + self + b0) ----
  gcn_gemm_wmma<128, 128><<<g128, 32, 0, stream>>>(x, W0, bufA);
  gcn_zero_f32<<<fin128Blk, B, 0, stream>>>(bufB, feat128);
  gcn_edge_scatter<128><<<scatBlocks, B, 0, stream>>>(bufA, src, dst, nrm, bufB, N_EDGES);
  gcn_finalize<128, true><<<fin128Blk, B, 0, stream>>>(bufB, bufA, dinv, b0, feat128);

  // ---- layer 1 ----
  gcn_gemm_wmma<128, 128><<<g128, 32, 0, stream>>>(bufB, W1, bufA);
  gcn_zero_f32<<<fin128Blk, B, 0, stream>>>(bufB, feat128);
  gcn_edge_scatter<128><<<scatBlocks, B, 0, stream>>>(bufA, src, dst, nrm, bufB, N_EDGES);
  gcn_finalize<128, true><<<fin128Blk, B, 0, stream>>>(bufB, bufA, dinv, b1, feat128);

  // ---- layer 2 (no relu) ----
  gcn_gemm_wmma<128, 64><<<g64, 32, 0, stream>>>(bufB, W2, bufA);
  gcn_zero_f32<<<fin64Blk, B, 0, stream>>>(bufB, feat64);
  gcn_edge_scatter<64><<<scatBlocks, B, 0, stream>>>(bufA, src, dst, nrm, bufB, N_EDGES);
  gcn_finalize<64, false><<<fin64Blk, B, 0, stream>>>(bufB, bufA, dinv, b2, feat64);

  // ---- mean pool + softmax ----
  gcn_pool<<<poolBlocks, B, 0, stream>>>(bufB, batch, sums, cnts, N_NODES);
  gcn_softmax<<<1, 128, 0, stream>>>(sums, cnts, out);
}